// GATLayer_1_12567074308557
// MI455X (gfx1250) — compile-verified
//
#include <hip/hip_runtime.h>
#include <math.h>

// ---------------------------------------------------------------------------
// CDNA5 (gfx1250) wave32 GAT implementation.
// Dense projections: V_WMMA_F32_16X16X4_F32, B staged in LDS (320KB/WGP),
// 8 waves/block, each wave = 16-row tile x all column tiles.
// ---------------------------------------------------------------------------

typedef __attribute__((ext_vector_type(2))) float v2f;
typedef __attribute__((ext_vector_type(4))) float v4f;
typedef __attribute__((ext_vector_type(8))) float v8f;

#define ORD_NEG_INF 0x007FFFFFu   // f2ord(-inf)

__device__ __forceinline__ unsigned f2ord(float f) {
  unsigned u = __float_as_uint(f);
  return (u & 0x80000000u) ? ~u : (u | 0x80000000u);
}
__device__ __forceinline__ float ord2f(unsigned u) {
  return (u & 0x80000000u) ? __uint_as_float(u & 0x7FFFFFFFu)
                           : __uint_as_float(~u);
}
__device__ __forceinline__ float leaky(float x) { return x > 0.f ? x : 0.2f * x; }

// --------------------------- generic fill ----------------------------------
__global__ void fill_u32(unsigned* p, unsigned v, size_t n) {
  size_t i = (size_t)blockIdx.x * blockDim.x + threadIdx.x;
  size_t st = (size_t)gridDim.x * blockDim.x;
  for (; i < n; i += st) p[i] = v;
}

// ------------------------ self-loop edge weight ----------------------------
__global__ void selfloop_accum(const int* ei, const float* ew, float* cnt,
                               float* lw, int E) {
  int e = blockIdx.x * blockDim.x + threadIdx.x;
  if (e >= E) return;
  int d = ei[E + e];
  atomicAdd(&cnt[d], 1.0f);
  atomicAdd(&lw[d], ew[e]);
}
__global__ void selfloop_div(float* lw, const float* cnt, int n) {
  int i = blockIdx.x * blockDim.x + threadIdx.x;
  if (i < n) lw[i] /= fmaxf(cnt[i], 1.0f);
}

// ------------------- WMMA f32 GEMM with LDS-staged B -----------------------
// C[M x NC] = A[M x K] * B[K x NC].  M must be a multiple of 16.
// Block: 256 threads = 8 waves. B (K*NC*4 <= 64KB) staged once into LDS.
// Wave w handles row-tile (blockIdx.x*8 + w): 16 rows x all NC columns.
// Per k-step (K in steps of 4): one global b64 A-load, then NC/16 WMMAs
// reading B pairs from LDS. No per-lane predication: EXEC all-ones for WMMA.
template <int NC, int K>
__global__ __launch_bounds__(256) void wmma_gemm_f32_lds(
    const float* __restrict__ A, const float* __restrict__ B,
    float* __restrict__ C, int M) {
  __shared__ float Bs[K * NC];        // 64KB (NC=128) / 32KB (NC=64)
  const int tid = threadIdx.x;

  // ---- cooperative stage of B into LDS (coalesced float4) ----
  constexpr int TOT4 = (K * NC) / 4;
  const v4f* B4 = (const v4f*)B;
  v4f* Bs4 = (v4f*)Bs;
#pragma unroll 4
  for (int i = tid; i < TOT4; i += 256) Bs4[i] = B4[i];
  __syncthreads();

  const int wave = tid >> 5;
  const int lane = tid & 31;
  const int half = lane >> 4;         // 0: K pair (k0,k0+1), 1: (k0+2,k0+3)
  const int l    = lane & 15;
  const int tile = blockIdx.x * 8 + wave;      // 16-row tile index
  if (tile * 16 >= M) return;                  // wave-uniform exit (post-barrier)
  const int rowBase = tile * 16;

  constexpr int CT = NC / 16;                  // column tiles per wave
  v8f acc[CT];
#pragma unroll
  for (int c = 0; c < CT; ++c) acc[c] = (v8f){};

  const float* ap = A + (size_t)(rowBase + l) * K + half * 2;
  for (int k0 = 0; k0 < K; k0 += 4) {
    const v2f a = *(const v2f*)(ap + k0);      // 8B-aligned (K even)
    const int ak = k0 + half * 2;
#pragma unroll
    for (int c = 0; c < CT; ++c) {
      v2f b;
      b.x = Bs[ak * NC + c * 16 + l];
      b.y = Bs[(ak + 1) * NC + c * 16 + l];
      acc[c] = __builtin_amdgcn_wmma_f32_16x16x4_f32(
          /*neg_a=*/false, a, /*neg_b=*/false, b,
          /*c_mod=*/(short)0, acc[c], /*reuse_a=*/false, /*reuse_b=*/false);
    }
  }

  // ---- store D: lane group selects column, vgpr index selects row ----
  const int row0 = rowBase + half * 8;
#pragma unroll
  for (int c = 0; c < CT; ++c) {
    const int col = c * 16 + l;
#pragma unroll
    for (int i = 0; i < 8; ++i)
      C[(size_t)(row0 + i) * NC + col] = acc[c][i];
  }
}

// ---------- per-node attention dots: als[n,h]=h·att_s, ald[n,h]=h·att_d ----
__global__ void node_dots(const float* __restrict__ hm,
                          const float* __restrict__ as_,
                          const float* __restrict__ ad_,
                          float* __restrict__ als, float* __restrict__ ald,
                          int n, int F, int Hh, int DHd) {
  const int wpb  = blockDim.x >> 5;
  const int node = blockIdx.x * wpb + (threadIdx.x >> 5);
  const int lane = threadIdx.x & 31;
  if (node >= n) return;
  float ps[2] = {0.f, 0.f}, pd[2] = {0.f, 0.f};
  for (int f = lane; f < F; f += 32) {
    const float v = hm[(size_t)node * F + f];
    const int h = f / DHd;
    ps[h] += v * as_[f];
    pd[h] += v * ad_[f];
  }
#pragma unroll
  for (int off = 16; off > 0; off >>= 1) {
    ps[0] += __shfl_xor(ps[0], off, 32);
    ps[1] += __shfl_xor(ps[1], off, 32);
    pd[0] += __shfl_xor(pd[0], off, 32);
    pd[1] += __shfl_xor(pd[1], off, 32);
  }
  if (lane == 0) {
    for (int h = 0; h < Hh; ++h) {
      als[(size_t)node * Hh + h] = ps[h];
      ald[(size_t)node * Hh + h] = pd[h];
    }
  }
}

// ---- c[h] = sum_d We[h*DH+d] * att_e[h*DH+d]   (rank-1 edge-embed dot) ----
__global__ void compute_c(const float* We, const float* ae, float* c,
                          int Hh, int DHd) {
  const int h = threadIdx.x;
  if (h < Hh) {
    float s = 0.f;
    for (int d = 0; d < DHd; ++d) s += We[h * DHd + d] * ae[h * DHd + d];
    c[h] = s;
  }
}

// --------------------- edge pass 1: segment max of logits ------------------
__global__ void edge_logit_max(const int* __restrict__ ei,
                               const float* __restrict__ ew,
                               const float* __restrict__ lw,
                               const float* __restrict__ als,
                               const float* __restrict__ ald,
                               const float* __restrict__ c,
                               unsigned* __restrict__ mOrd,
                               int E, int n, int Hh) {
  const int E2 = E + n;
  const int e = blockIdx.x * blockDim.x + threadIdx.x;
  if (e >= E2) return;
  int s, d; float w;
  if (e < E) { s = ei[e]; d = ei[E + e]; w = ew[e]; }
  else       { s = e - E; d = s;         w = lw[s]; }
  for (int h = 0; h < Hh; ++h) {
    const float lg = leaky(als[(size_t)s * Hh + h] + ald[(size_t)d * Hh + h] + w * c[h]);
    atomicMax(&mOrd[(size_t)d * Hh + h], f2ord(lg));
  }
}

// ------------- edge pass 2: ex = exp(logit - m), den = segsum(ex) ----------
__global__ void edge_exp_den(const int* __restrict__ ei,
                             const float* __restrict__ ew,
                             const float* __restrict__ lw,
                             const float* __restrict__ als,
                             const float* __restrict__ ald,
                             const float* __restrict__ c,
                             const unsigned* __restrict__ mOrd,
                             float* __restrict__ ex, float* __restrict__ den,
                             int E, int n, int Hh) {
  const int E2 = E + n;
  const int e = blockIdx.x * blockDim.x + threadIdx.x;
  if (e >= E2) return;
  int s, d; float w;
  if (e < E) { s = ei[e]; d = ei[E + e]; w = ew[e]; }
  else       { s = e - E; d = s;         w = lw[s]; }
  for (int h = 0; h < Hh; ++h) {
    const float lg = leaky(als[(size_t)s * Hh + h] + ald[(size_t)d * Hh + h] + w * c[h]);
    const float m  = ord2f(mOrd[(size_t)d * Hh + h]);
    const float x  = expf(lg - m);
    ex[(size_t)e * Hh + h] = x;
    atomicAdd(&den[(size_t)d * Hh + h], x);
  }
}

// ---- edge pass 3: out[dst] += h[src] * alpha   (block = F feature lanes) --
__global__ void edge_scatter(const int* __restrict__ ei,
                             const float* __restrict__ hm,
                             const float* __restrict__ ex,
                             const float* __restrict__ den,
                             float* __restrict__ out,
                             int E, int n, int Hh, int DHd) {
  const int F = Hh * DHd;
  const int e = blockIdx.x;
  const int t = threadIdx.x;
  int s, d;
  if (e < E) { s = ei[e]; d = ei[E + e]; }
  else       { s = e - E; d = s; }
  const int h = t / DHd;
  const float alpha = ex[(size_t)e * Hh + h] / den[(size_t)d * Hh + h];
  atomicAdd(&out[(size_t)d * F + t], hm[(size_t)s * F + t] * alpha);
}

// ------------------------ epilogues ----------------------------------------
__global__ void bias_relu(float* o, const float* b, size_t total, int F) {
  size_t i = (size_t)blockIdx.x * blockDim.x + threadIdx.x;
  size_t st = (size_t)gridDim.x * blockDim.x;
  for (; i < total; i += st) o[i] = fmaxf(o[i] + b[i % F], 0.f);
}
__global__ void bias_add(float* o, const float* b, size_t total, int F) {
  size_t i = (size_t)blockIdx.x * blockDim.x + threadIdx.x;
  size_t st = (size_t)gridDim.x * blockDim.x;
  for (; i < total; i += st) o[i] = o[i] + b[i % F];
}

// ---------------------------------------------------------------------------
extern "C" void kernel_launch(void* const* d_in, const int* in_sizes, int n_in,
                              void* d_out, int out_size, void* d_ws, size_t ws_size,
                              hipStream_t stream) {
  const float* x   = (const float*)d_in[0];
  const int*   ei  = (const int*)  d_in[1];
  const float* ew  = (const float*)d_in[2];
  const float* W1  = (const float*)d_in[3];
  const float* as1 = (const float*)d_in[4];
  const float* ad1 = (const float*)d_in[5];
  const float* ae1 = (const float*)d_in[6];
  const float* We1 = (const float*)d_in[7];
  const float* b1  = (const float*)d_in[8];
  const float* W2  = (const float*)d_in[9];
  const float* as2 = (const float*)d_in[10];
  const float* ad2 = (const float*)d_in[11];
  const float* ae2 = (const float*)d_in[12];
  const float* We2 = (const float*)d_in[13];
  const float* b2  = (const float*)d_in[14];

  const int DIN = 128, H1 = 2, DH1 = 64, F1 = H1 * DH1;  // = 128
  const int H2 = 1, DH2 = 64, F2 = 64;                   // DOUT
  const int n  = in_sizes[0] / DIN;    // 50000 (multiple of 16)
  const int E  = in_sizes[1] / 2;      // 800000
  const int E2 = E + n;

  // ------------------------- workspace layout (floats) --------------------
  float* ws = (float*)d_ws;
  size_t o = 0;
  float*    cnt   = ws + o;             o += (size_t)n;
  float*    lw    = ws + o;             o += (size_t)n;         // self-loop weights
  float*    h1    = ws + o;             o += (size_t)n * F1;
  float*    als1  = ws + o;             o += (size_t)n * H1;
  float*    ald1  = ws + o;             o += (size_t)n * H1;
  unsigned* mOrd1 = (unsigned*)(ws+o);  o += (size_t)n * H1;
  float*    den1  = ws + o;             o += (size_t)n * H1;
  float*    ex1   = ws + o;             o += (size_t)E2 * H1;
  float*    out1  = ws + o;             o += (size_t)n * F1;    // becomes relu(h)
  float*    h2    = ws + o;             o += (size_t)n * F2;
  float*    als2  = ws + o;             o += (size_t)n;
  float*    ald2  = ws + o;             o += (size_t)n;
  unsigned* mOrd2 = (unsigned*)(ws+o);  o += (size_t)n;
  float*    den2  = ws + o;             o += (size_t)n;
  float*    ex2   = ws + o;             o += (size_t)E2;
  float*    c1    = ws + o;             o += 2;
  float*    c2    = ws + o;             o += 2;
  float*    outf  = (float*)d_out;      // N x 64 accumulator, then +b2

  const int TB = 256;
  auto blocks = [](size_t total, int tb) {
    size_t b = (total + tb - 1) / tb;
    return (int)(b > 65535u * 16u ? 65535u * 16u : b);
  };

  // ------------------------- zero / init accumulators ---------------------
  fill_u32<<<blocks((size_t)n * 2, TB), TB, 0, stream>>>((unsigned*)cnt, 0u, (size_t)n * 2);
  fill_u32<<<blocks((size_t)n * H1, TB), TB, 0, stream>>>((unsigned*)den1, 0u, (size_t)n * H1);
  fill_u32<<<blocks((size_t)n * F1, TB), TB, 0, stream>>>((unsigned*)out1, 0u, (size_t)n * F1);
  fill_u32<<<blocks((size_t)n, TB), TB, 0, stream>>>((unsigned*)den2, 0u, (size_t)n);
  fill_u32<<<blocks((size_t)n * F2, TB), TB, 0, stream>>>((unsigned*)outf, 0u, (size_t)n * F2);
  fill_u32<<<blocks((size_t)n * H1, TB), TB, 0, stream>>>(mOrd1, ORD_NEG_INF, (size_t)n * H1);
  fill_u32<<<blocks((size_t)n, TB), TB, 0, stream>>>(mOrd2, ORD_NEG_INF, (size_t)n);

  // ------------------------- self-loop edge weights -----------------------
  selfloop_accum<<<(E + TB - 1) / TB, TB, 0, stream>>>(ei, ew, cnt, lw, E);
  selfloop_div<<<(n + TB - 1) / TB, TB, 0, stream>>>(lw, cnt, n);

  const int rowTiles = (n + 15) / 16;          // 3125
  const int gemmBlocks = (rowTiles + 7) / 8;   // 8 waves (row tiles) per block

  // =========================== layer 1 ====================================
  wmma_gemm_f32_lds<128, 128><<<gemmBlocks, 256, 0, stream>>>(x, W1, h1, n);
  node_dots<<<(n * 32 + TB - 1) / TB, TB, 0, stream>>>(h1, as1, ad1, als1, ald1, n, F1, H1, DH1);
  compute_c<<<1, H1, 0, stream>>>(We1, ae1, c1, H1, DH1);
  edge_logit_max<<<(E2 + TB - 1) / TB, TB, 0, stream>>>(ei, ew, lw, als1, ald1, c1, mOrd1, E, n, H1);
  edge_exp_den<<<(E2 + TB - 1) / TB, TB, 0, stream>>>(ei, ew, lw, als1, ald1, c1, mOrd1, ex1, den1, E, n, H1);
  edge_scatter<<<E2, F1, 0, stream>>>(ei, h1, ex1, den1, out1, E, n, H1, DH1);
  bias_relu<<<blocks((size_t)n * F1, TB), TB, 0, stream>>>(out1, b1, (size_t)n * F1, F1);

  // =========================== layer 2 ====================================
  wmma_gemm_f32_lds<64, 128><<<gemmBlocks, 256, 0, stream>>>(out1, W2, h2, n);
  node_dots<<<(n * 32 + TB - 1) / TB, TB, 0, stream>>>(h2, as2, ad2, als2, ald2, n, F2, H2, DH2);
  compute_c<<<1, H2, 0, stream>>>(We2, ae2, c2, H2, DH2);
  edge_logit_max<<<(E2 + TB - 1) / TB, TB, 0, stream>>>(ei, ew, lw, als2, ald2, c2, mOrd2, E, n, H2);
  edge_exp_den<<<(E2 + TB - 1) / TB, TB, 0, stream>>>(ei, ew, lw, als2, ald2, c2, mOrd2, ex2, den2, E, n, H2);
  edge_scatter<<<E2, F2, 0, stream>>>(ei, h2, ex2, den2, outf, E, n, H2, DH2);
  bias_add<<<blocks((size_t)n * F2, TB), TB, 0, stream>>>(outf, b2, (size_t)n * F2, F2);
}